// LearnableTensorSquare_34711925686401
// MI455X (gfx1250) — compile-verified
//
#include <hip/hip_runtime.h>

typedef __attribute__((ext_vector_type(2))) float v2f;
typedef __attribute__((ext_vector_type(8))) float v8f;

#define ZDIM 4096
#define DI   256              // dim_in (i and j)
#define NT   2048             // dim_ts (t)
#define DOUT 256              // dim_out (k)
#define NN   (DI * DI)        // 65536, the fused ij dimension

// ============================================================================
// Stage 1:  Kmat[o, n] = sum_t W[o, t] * T[t, n]
//   M=256 (o), N=65536 (n), K=2048 (t); f32 WMMA 16x16x4.
//   One block owns ALL 256 o-rows for a 64-column n stripe -> T is read from
//   HBM exactly once (512 MB @ 23.3 TB/s ~= 22 us lower bound).
//   Wave w owns o-rows [32w, 32w+32): 2 o-tiles x 4 n-tiles = 8 accum tiles.
// ============================================================================
__global__ __launch_bounds__(256) void stage1_WT(const float* __restrict__ W,
                                                 const float* __restrict__ T,
                                                 float* __restrict__ Kmat) {
  const int lane = threadIdx.x & 31;
  const int wave = threadIdx.x >> 5;   // 0..7
  const int lo   = lane & 15;
  const int hi   = lane >> 4;          // 0 or 1

  const int nb = blockIdx.x * 64;      // n stripe base
  const int ob = wave * 32;            // o base for this wave

  const v8f vzero = {0.f, 0.f, 0.f, 0.f, 0.f, 0.f, 0.f, 0.f};
  v8f c[2][4];
#pragma unroll
  for (int i = 0; i < 2; ++i)
#pragma unroll
    for (int j = 0; j < 4; ++j) c[i][j] = vzero;

  // A-frag (16x4 f32 layout): lane<16 holds K=t,t+1 ; lane>=16 holds K=t+2,t+3
  const float* Arow0 = W + (size_t)(ob + lo) * NT + 2 * hi;
  const float* Arow1 = W + (size_t)(ob + 16 + lo) * NT + 2 * hi;
  // B-frag (4x16): lane<16 cols n, rows t+0/t+1 ; lane>=16 rows t+2/t+3
  const float* Bcol = T + (size_t)(2 * hi) * NN + nb + lo;

  for (int t = 0; t < NT; t += 4) {
    // speculative prefetch of the T stream into GL2 (OOB tail silently dropped)
    __builtin_prefetch(Bcol + (size_t)(t + 8) * NN, 0, 0);

    v2f a0 = *(const v2f*)(Arow0 + t);
    v2f a1 = *(const v2f*)(Arow1 + t);
#pragma unroll
    for (int nt = 0; nt < 4; ++nt) {
      const float* bp = Bcol + (size_t)t * NN + nt * 16;
      v2f b;
      b.x = bp[0];
      b.y = bp[NN];
      c[0][nt] = __builtin_amdgcn_wmma_f32_16x16x4_f32(false, a0, false, b,
                                                       (short)0, c[0][nt], false, false);
      c[1][nt] = __builtin_amdgcn_wmma_f32_16x16x4_f32(false, a1, false, b,
                                                       (short)0, c[1][nt], false, false);
    }
  }

  // D layout: VGPR v -> row (v) for lanes 0-15, row (v+8) for lanes 16-31
#pragma unroll
  for (int ot = 0; ot < 2; ++ot)
#pragma unroll
    for (int nt = 0; nt < 4; ++nt)
#pragma unroll
      for (int v = 0; v < 8; ++v) {
        const int o = ob + ot * 16 + v + 8 * hi;
        const int n = nb + nt * 16 + lo;
        Kmat[(size_t)o * NN + n] = c[ot][nt][v];
      }
}

// ============================================================================
// Stage 2:  out[z, k] = sum_n M[z, n] * Kmat[k, n],  M[z, i*256+j] = x[z,i]*x[z,j]
//   A (M) is generated in-register from a 64x256 x-tile staged in 64 KB LDS.
//   LDS layout keeps (j, j+1) pairs contiguous per z-row with an XOR bank
//   swizzle, so each lane fetches both j-factors with ONE ds_load_b64:
//       addr(j, zl) = (j>>1)*128 + 2*(zl ^ ((j>>1)&63)) + (j&1)
//   The n-loop is split i (outer) x jj (inner): x[z,i] hoists out of the
//   inner loop and pair loads never cross an i-row.
//   B-frags are contiguous float2 loads from Kmat (L2-resident, 64 MB).
//   n is split gridDim.y ways into private partials (deterministic).
//   Block: 8 waves = 2 z-groups x 4 k-groups; wave tile 32z x 64k.
// ============================================================================
__global__ __launch_bounds__(256) void stage2_out(const float* __restrict__ X,
                                                  const float* __restrict__ Kmat,
                                                  float* __restrict__ outBase,
                                                  int outStride, int nPerSplit) {
  __shared__ float xs[64 * 256];  // pair-contiguous, XOR-swizzled

  const int tid  = threadIdx.x;
  const int lane = tid & 31;
  const int wave = tid >> 5;
  const int lo   = lane & 15;
  const int hi   = lane >> 4;
  const int zg   = wave >> 2;     // 0..1  (z-group of 32 rows)
  const int kg   = wave & 3;      // 0..3  (k-group of 64 cols)

  const int z0   = blockIdx.x * 64;
  const int ibeg = (blockIdx.y * nPerSplit) >> 8;   // nPerSplit is a multiple of 256
  const int iend = ibeg + (nPerSplit >> 8);
  float* out = outBase + (size_t)blockIdx.y * outStride;

  // cooperative, coalesced fill of the x tile into swizzled LDS
  for (int idx = tid; idx < 64 * 256; idx += 256) {
    const int zl = idx >> 8;
    const int j  = idx & 255;
    const int p  = j >> 1;
    xs[p * 128 + 2 * (zl ^ (p & 63)) + (j & 1)] = X[(size_t)(z0 + zl) * DI + j];
  }
  __syncthreads();

  const v8f vzero = {0.f, 0.f, 0.f, 0.f, 0.f, 0.f, 0.f, 0.f};
  v8f c[2][4];
#pragma unroll
  for (int i = 0; i < 2; ++i)
#pragma unroll
    for (int j = 0; j < 4; ++j) c[i][j] = vzero;

  const int zl0 = zg * 32 + lo;      // z-tile 0 lane row (local)
  const int zl1 = zl0 + 16;          // z-tile 1 lane row
  // B lane row pointer: k = kg*64 + kt*16 + lo ; +2*hi selects this lane's n pair
  const float* BpLane = Kmat + (size_t)(kg * 64 + lo) * NN + 2 * hi;

  for (int i = ibeg; i < iend; ++i) {
    const int pi = i >> 1;
    const float xi0 = xs[pi * 128 + 2 * (zl0 ^ (pi & 63)) + (i & 1)];
    const float xi1 = xs[pi * 128 + 2 * (zl1 ^ (pi & 63)) + (i & 1)];
    const float* Bpi = BpLane + (size_t)i * DI;

    for (int jj = 0; jj < DI; jj += 4) {
      const int p = (jj >> 1) + hi;               // pair index of this lane's ja
      v2f xj0 = *(const v2f*)(xs + p * 128 + 2 * (zl0 ^ (p & 63)));
      v2f xj1 = *(const v2f*)(xs + p * 128 + 2 * (zl1 ^ (p & 63)));
      v2f a0, a1;
      a0.x = xi0 * xj0.x;
      a0.y = xi0 * xj0.y;
      a1.x = xi1 * xj1.x;
      a1.y = xi1 * xj1.y;

#pragma unroll
      for (int kt = 0; kt < 4; ++kt) {
        v2f b = *(const v2f*)(Bpi + (size_t)kt * 16 * NN + jj);  // contiguous pair
        c[0][kt] = __builtin_amdgcn_wmma_f32_16x16x4_f32(false, a0, false, b,
                                                         (short)0, c[0][kt], false, false);
        c[1][kt] = __builtin_amdgcn_wmma_f32_16x16x4_f32(false, a1, false, b,
                                                         (short)0, c[1][kt], false, false);
      }
    }
  }

#pragma unroll
  for (int zt = 0; zt < 2; ++zt)
#pragma unroll
    for (int kt = 0; kt < 4; ++kt)
#pragma unroll
      for (int v = 0; v < 8; ++v) {
        const int z = z0 + zg * 32 + zt * 16 + v + 8 * hi;
        const int k = kg * 64 + kt * 16 + lo;
        out[(size_t)z * DOUT + k] = c[zt][kt][v];
      }
}

// ============================================================================
// Reduce the 4 n-split partials into d_out.
// ============================================================================
__global__ __launch_bounds__(256) void reduce4(const float* __restrict__ p,
                                               float* __restrict__ out,
                                               int n, int stride) {
  const int i = blockIdx.x * blockDim.x + threadIdx.x;
  if (i < n) {
    out[i] = p[i] + p[i + stride] + p[i + 2 * stride] + p[i + 3 * stride];
  }
}

extern "C" void kernel_launch(void* const* d_in, const int* in_sizes, int n_in,
                              void* d_out, int out_size, void* d_ws, size_t ws_size,
                              hipStream_t stream) {
  const float* X = (const float*)d_in[0];  // features [4096, 256]
  const float* T = (const float*)d_in[1];  // [2048, 65536]
  const float* W = (const float*)d_in[2];  // [256, 2048]
  float* out = (float*)d_out;              // [4096, 256]

  float* Kmat = (float*)d_ws;              // 64 MB: [256, 65536] f32
  const size_t kElems = (size_t)DOUT * NN;

  // Stage 1: K = W @ T  (1024 blocks x 256 threads)
  stage1_WT<<<NN / 64, 256, 0, stream>>>(W, T, Kmat);

  const int zk = ZDIM * DOUT;  // 1M output elements
  const size_t needSplit = (kElems + 4 * (size_t)zk) * sizeof(float);  // 80 MB

  if (ws_size >= needSplit) {
    // 4-way n split -> 64x4 = 256 blocks; partials in scratch, then reduce.
    float* part = Kmat + kElems;
    stage2_out<<<dim3(ZDIM / 64, 4), 256, 0, stream>>>(X, Kmat, part, zk, NN / 4);
    reduce4<<<zk / 256, 256, 0, stream>>>(part, out, zk, zk);
  } else {
    // Fallback: single pass over n, write d_out directly.
    stage2_out<<<dim3(ZDIM / 64, 1), 256, 0, stream>>>(X, Kmat, out, 0, NN);
  }
}